// PixelConv_89970974916755
// MI455X (gfx1250) — compile-verified
//
#include <hip/hip_runtime.h>
#include <stdint.h>
#include <stddef.h>

// ---------------------------------------------------------------------------
// PixelConv (scale=2, depthwise, 3x3) for MI455X / gfx1250.
//
// Roofline: 472 MFLOP vs ~161 MB moved -> ~3 FLOP/byte -> bandwidth bound
// (~6.9 us floor at 23.3 TB/s). WMMA 16x16x4_f32 cannot express the N=4,K=9
// per-pixel contraction without >=75% tile waste, so the optimal path is
// VALU f32 FMA fed by the Tensor Data Mover: TDM stages the 32x-reused
// per-pixel weights and the 9x-reused feature halo rows in LDS, and the
// 105 MB output stream goes out through non-temporal b64 stores.
// ---------------------------------------------------------------------------

#if defined(__has_builtin)
#if __has_builtin(__builtin_amdgcn_tensor_load_to_lds)
#define USE_TDM 1
#else
#define USE_TDM 0
#endif
#else
#define USE_TDM 0
#endif

typedef __attribute__((ext_vector_type(4))) unsigned int v4u_t;
typedef __attribute__((ext_vector_type(8))) int v8i_t;
typedef __attribute__((ext_vector_type(4))) int v4i_t;
typedef __attribute__((ext_vector_type(2))) float v2f_t;

typedef __attribute__((address_space(3))) void lds_void_t;

__device__ static inline uint32_t lds_byte_off(void* p) {
  // generic -> LDS addrspace, then ptrtoint gives the LDS byte offset
  return (uint32_t)(uintptr_t)(lds_void_t*)p;
}

#define DIM_H 160
#define DIM_W 160
#define DIM_C 32
#define DIM_N 8
#define CCHUNK 16

#if USE_TDM
// Issue one TDM load. Group0: count=1(valid), lds byte addr, 57-bit global
// byte addr, type=2 ("image"). Groups per CDNA5 ISA 8.3-8.6. This toolchain's
// builtin takes 6 args: (v4u g0, v8i g1, v4i g2, v4i g3, v8i extra, i32 cpol);
// the trailing v8i is uncharacterized (reserved/VADDR4) -> zeros.
__device__ static inline void tdm_issue(uint32_t lds_addr, uint64_t gaddr,
                                        v8i_t g1, v4i_t g2) {
  v4u_t g0 = { 1u,
               lds_addr,
               (uint32_t)gaddr,
               (uint32_t)((gaddr >> 32) & 0x01FFFFFFu) | (2u << 30) };
  v4i_t g3 = { 0, 0, 0, 0 };
  v8i_t g4 = { 0, 0, 0, 0, 0, 0, 0, 0 };
  __builtin_amdgcn_tensor_load_to_lds(g0, g1, g2, g3, g4, 0);
}

// Feature chunk: 2-D tile (W x nrows), iterated CCHUNK times over channels
// (iterate_enable: lds += 3*W elems, global += H*W elems per iteration).
__device__ static inline void tdm_feat_chunk(const float* feat, float* fTile,
                                             int n, int cbase, int row0,
                                             int nrows, int skip) {
  uint64_t ga = (uint64_t)(uintptr_t)(feat +
      ((size_t)(n * DIM_C + cbase) * (DIM_H * DIM_W) + (size_t)row0 * DIM_W));
  v8i_t g1 = { (int)((2u << 16) | (1u << 19)),   // data_size=4B, iterate_enable
               (int)(DIM_W << 16),               // tensor_dim0 = 160
               (int)(nrows << 16),               // tensor_dim1 = nrows
               (int)(DIM_W << 16),               // tile_dim0 = 160
               nrows,                            // tile_dim1 = nrows
               DIM_W,                            // tensor_dim0_stride = 160
               0, 0 };
  v4i_t g2 = { 0,
               3 * DIM_W,                        // lds_addr_increment (elems)
               DIM_H * DIM_W,                    // global_addr_increment (elems)
               (int)((CCHUNK - 1) << 16) };      // iterate_count: 15 -> 16 iters
  tdm_issue(lds_byte_off(fTile) + (uint32_t)(skip * DIM_W * 4), ga, g1, g2);
}
#else
__device__ static inline void copy_feat_chunk(const float* feat, float* fTile,
                                              int n, int cbase, int row0,
                                              int nrows, int skip, int tid) {
  for (int i = tid; i < CCHUNK * nrows * DIM_W; i += 160) {
    int x = i % DIM_W;
    int t = i / DIM_W;
    int j = t % nrows;
    int c = t / nrows;
    fTile[c * (3 * DIM_W) + (j + skip) * DIM_W + x] =
        feat[(size_t)(n * DIM_C + cbase + c) * (DIM_H * DIM_W) +
             (size_t)(row0 + j) * DIM_W + x];
  }
}
#endif

__global__ __launch_bounds__(160) void
pixelconv_tdm_kernel(const float* __restrict__ feat,
                     const float* __restrict__ kern,
                     float* __restrict__ out) {
  constexpr int H = DIM_H, W = DIM_W, C = DIM_C;
  constexpr int H2 = 2 * H, W2 = 2 * W;

  __shared__ float kTile[36 * DIM_W];           // 23040 B: per-pixel weights
  __shared__ float fTile[CCHUNK * 3 * DIM_W];   // 30720 B: 16 ch x 3 rows

  const int h = blockIdx.x;
  const int n = blockIdx.y;
  const int tid = (int)threadIdx.x;
  const int lane = tid & 31;
  const int wave = tid >> 5;
  const int w = wave * 32 + lane;               // one w-strip per wave32

  const int row0  = (h == 0) ? 0 : (h - 1);
  const int nrows = (h == 0 || h == H - 1) ? 2 : 3;
  const int skip  = (h == 0) ? 1 : 0;

  // Pre-zero the missing halo row slot at image top/bottom; TDM never writes
  // that slot, so the zeros survive both channel chunks.
  if (nrows == 2) {
    const int slot = (h == 0) ? 0 : 2;
    for (int c = 0; c < CCHUNK; ++c)
      fTile[c * (3 * W) + slot * W + w] = 0.0f;
  }

#if USE_TDM
  if (tid < 32) {   // wave 0 drives the DMA
    {
      // weight tile: 36 planes x 160 w, plane stride H*W
      uint64_t ga = (uint64_t)(uintptr_t)(kern +
          ((size_t)n * 36 * (H * W) + (size_t)h * W));
      v8i_t g1 = { (int)(2u << 16),      // data_size=4B
                   (int)(W << 16),       // tensor_dim0 = 160
                   (int)(36 << 16),      // tensor_dim1 = 36
                   (int)(W << 16),       // tile_dim0 = 160
                   36,                   // tile_dim1 = 36
                   H * W,                // tensor_dim0_stride = 25600
                   0, 0 };
      v4i_t g2 = { 0, 0, 0, 0 };
      tdm_issue(lds_byte_off(kTile), ga, g1, g2);
    }
    tdm_feat_chunk(feat, fTile, n, 0, row0, nrows, skip);
    __builtin_amdgcn_s_wait_tensorcnt(0);
  }
#else
  for (int i = tid; i < 36 * W; i += 160)
    kTile[i] = kern[(size_t)n * 36 * (H * W) + (size_t)(i / W) * (H * W) +
                    (size_t)h * W + (i % W)];
  copy_feat_chunk(feat, fTile, n, 0, row0, nrows, skip, tid);
#endif
  __syncthreads();

  // Per-strip weights, loaded once and reused for all 32 channels.
  // Left/right image-border taps are folded in by zeroing the weights.
  float wk[36];
  int xs[3];
#pragma unroll
  for (int dx = 0; dx < 3; ++dx) {
    int x = w + dx - 1;
    bool ok = (unsigned)x < (unsigned)W;
    xs[dx] = ok ? x : ((x < 0) ? 0 : W - 1);
#pragma unroll
    for (int j = 0; j < 3; ++j) {
      const int k = dx * 3 + j;
#pragma unroll
      for (int s = 0; s < 4; ++s)
        wk[k * 4 + s] = ok ? kTile[(k * 4 + s) * W + w] : 0.0f;
    }
  }

  float* outBase =
      out + (((size_t)(n * C) * H2 + (size_t)(2 * h)) * W2 + (size_t)(2 * w));
  const size_t cstride = (size_t)H2 * W2;

  for (int phase = 0; phase < 2; ++phase) {
    if (phase == 1) {
      __syncthreads();          // everyone done reading chunk 0
#if USE_TDM
      if (tid < 32) {
        tdm_feat_chunk(feat, fTile, n, CCHUNK, row0, nrows, skip);
        __builtin_amdgcn_s_wait_tensorcnt(0);
      }
#else
      copy_feat_chunk(feat, fTile, n, CCHUNK, row0, nrows, skip, tid);
#endif
      __syncthreads();
    }
    const int cbase = phase * CCHUNK;
    for (int c = 0; c < CCHUNK; ++c) {
      const float* f0 = fTile + c * (3 * W);
      float a0 = 0.f, a1 = 0.f, a2 = 0.f, a3 = 0.f;
#pragma unroll
      for (int dx = 0; dx < 3; ++dx) {
#pragma unroll
        for (int j = 0; j < 3; ++j) {
          const float v = f0[j * W + xs[dx]];   // ds_load_b32, imm offset j*640
          const int k = dx * 3 + j;
          a0 = fmaf(v, wk[k * 4 + 0], a0);
          a1 = fmaf(v, wk[k * 4 + 1], a1);
          a2 = fmaf(v, wk[k * 4 + 2], a2);
          a3 = fmaf(v, wk[k * 4 + 3], a3);
        }
      }
      // out[n, c, 2h+r, 2w+q] = acc[r*2+q]; two adjacent q are contiguous.
      float* op = outBase + (size_t)(cbase + c) * cstride;
      v2f_t lo = { a0, a1 };
      v2f_t hi = { a2, a3 };
      __builtin_nontemporal_store(lo, (v2f_t*)op);
      __builtin_nontemporal_store(hi, (v2f_t*)(op + W2));
    }
  }
}

extern "C" void kernel_launch(void* const* d_in, const int* in_sizes, int n_in,
                              void* d_out, int out_size, void* d_ws,
                              size_t ws_size, hipStream_t stream) {
  (void)in_sizes; (void)n_in; (void)out_size; (void)d_ws; (void)ws_size;
  const float* feat = (const float*)d_in[0];   // [8, 32, 160, 160]
  const float* kern = (const float*)d_in[1];   // [8, 36, 160, 160]
  float* out = (float*)d_out;                  // [8, 32, 320, 320]
  dim3 grid(DIM_H, DIM_N);                     // one workgroup per (n, h) row
  pixelconv_tdm_kernel<<<grid, 160, 0, stream>>>(feat, kern, out);
}